// GCNLayer_31473520345935
// MI455X (gfx1250) — compile-verified
//
#include <hip/hip_runtime.h>
#include <hip/hip_bf16.h>

typedef float v2f __attribute__((ext_vector_type(2)));
typedef float v8f __attribute__((ext_vector_type(8)));

#define FEAT 64
#define SA 68   // LDS row stride (floats) for A staging: banks (4m+k)%64 -> conflict-free b64 pairs
#define S2 48   // LDS row stride (float2) for W pairs: 96 dwords = 32 mod 64 -> halves hit disjoint banks

// ---------------- degree: cnt[dst]++ ----------------
__global__ void gcn_deg_kernel(const int* __restrict__ ei, unsigned* __restrict__ cnt, int E) {
    int i = blockIdx.x * blockDim.x + threadIdx.x;
    if (i < E) {
        int d = ei[E + i];          // row 1 = dst
        atomicAdd(&cnt[d], 1u);
    }
}

// ---------------- dis = (deg+1)^-1/2 ----------------
__global__ void gcn_dis_kernel(const unsigned* __restrict__ cnt, float* __restrict__ dis, int N) {
    int i = blockIdx.x * blockDim.x + threadIdx.x;
    if (i < N) {
        dis[i] = rsqrtf((float)cnt[i] + 1.0f);
    }
}

// ---------------- scatter: agg[dst] += x[src]*dis[src] ----------------
// 16 threads per edge, each handles 4 contiguous features (float4 gather).
__global__ void gcn_scatter_kernel(const float* __restrict__ x, const int* __restrict__ ei,
                                   const float* __restrict__ dis, float* __restrict__ agg, int E) {
    int gid = blockIdx.x * blockDim.x + threadIdx.x;
    int e = gid >> 4;
    if (e >= E) return;
    int c = (gid & 15) << 2;
    int s = ei[e];                  // row 0 = src
    int d = ei[E + e];              // row 1 = dst
    float ds = dis[s];
    const float4 v = *(const float4*)(x + (size_t)s * FEAT + c);
    float* ap = agg + (size_t)d * FEAT + c;
    unsafeAtomicAdd(ap + 0, v.x * ds);
    unsafeAtomicAdd(ap + 1, v.y * ds);
    unsafeAtomicAdd(ap + 2, v.z * ds);
    unsafeAtomicAdd(ap + 3, v.w * ds);
}

// ---------------- fused normalize + GEMM via V_WMMA_F32_16X16X4_F32 ----------------
// out[n,o] = sum_k ((agg[n,k] + x[n,k]*dis[n]) * dis[n]) * W[o,k]
// Block: 64 threads = 2 waves, 32 nodes per block (N=100000 -> 3125 blocks exactly).
__global__ __launch_bounds__(64) void gcn_gemm_kernel(const float* __restrict__ agg,
                                                      const float* __restrict__ x,
                                                      const float* __restrict__ dis,
                                                      const float* __restrict__ W,
                                                      float* __restrict__ out, int N) {
    __shared__ float lds_a[32 * SA];          // a' staged, padded rows
    __shared__ float lds_wp[32 * S2 * 2];     // B pairs: row r=(kb4*2+half) holds {W[o][2r],W[o][2r+1]} per o

    const int tid = threadIdx.x;              // 0..63
    const int node0 = blockIdx.x * 32;

    // Stage W as interleaved B-fragment pairs: thread t owns output column o = t.
    {
        const int o = tid;
        const float* wrow = W + o * FEAT;
        #pragma unroll
        for (int r = 0; r < 32; ++r) {
            float* p = lds_wp + (r * S2 + o) * 2;
            p[0] = wrow[2 * r];
            p[1] = wrow[2 * r + 1];
        }
    }
    // Stage a' = (agg + x*dis)*dis for 32 nodes: thread handles half a node row.
    {
        const int n = tid >> 1;
        const int h = (tid & 1) * 32;
        int node = node0 + n;
        if (node > N - 1) node = N - 1;       // clamp (no divergence before WMMA)
        const float ds = dis[node];
        const float* xr = x + (size_t)node * FEAT + h;
        const float* ar = agg + (size_t)node * FEAT + h;
        float* lr = lds_a + n * SA + h;
        #pragma unroll
        for (int j = 0; j < 32; j += 4) {
            const float4 xv = *(const float4*)(xr + j);
            const float4 av = *(const float4*)(ar + j);
            float4 r;
            r.x = (av.x + xv.x * ds) * ds;
            r.y = (av.y + xv.y * ds) * ds;
            r.z = (av.z + xv.z * ds) * ds;
            r.w = (av.w + xv.w * ds) * ds;
            *(float4*)(lr + j) = r;
        }
    }
    __syncthreads();

    const int wave = tid >> 5;                // 0 or 1 : 16-node tile within block
    const int lane = tid & 31;
    const int lhalf = lane >> 4;              // 0: lanes 0-15, 1: lanes 16-31
    const int l15 = lane & 15;
    const int mrow0 = wave * 16;

    v8f c0 = {}, c1 = {}, c2 = {}, c3 = {};

    const float* abase = lds_a + (mrow0 + l15) * SA + lhalf * 2;
    const float* wbase = lds_wp + (lhalf * S2 + l15) * 2;

    #pragma unroll 4
    for (int kb4 = 0; kb4 < 16; ++kb4) {
        // A fragment: lanes 0-15 hold A[m][kb..kb+1], lanes 16-31 hold A[m][kb+2..kb+3]
        const v2f a = *(const v2f*)(abase + kb4 * 4);
        // B fragments: one aligned b64 load each; row (kb4*2 + lhalf), col obase+l15
        const float* wrow = wbase + kb4 * (2 * S2 * 2);
        const v2f b0 = *(const v2f*)(wrow + 0);
        const v2f b1 = *(const v2f*)(wrow + 32);
        const v2f b2 = *(const v2f*)(wrow + 64);
        const v2f b3 = *(const v2f*)(wrow + 96);
        c0 = __builtin_amdgcn_wmma_f32_16x16x4_f32(false, a, false, b0, (short)0, c0, false, false);
        c1 = __builtin_amdgcn_wmma_f32_16x16x4_f32(false, a, false, b1, (short)0, c1, false, false);
        c2 = __builtin_amdgcn_wmma_f32_16x16x4_f32(false, a, false, b2, (short)0, c2, false, false);
        c3 = __builtin_amdgcn_wmma_f32_16x16x4_f32(false, a, false, b3, (short)0, c3, false, false);
    }

    // Store C: VGPR r holds row r (lanes 0-15) / row r+8 (lanes 16-31), col = l15.
    const int nodebase = node0 + mrow0;
    #pragma unroll
    for (int r = 0; r < 8; ++r) {
        const int m = r + lhalf * 8;
        const int node = nodebase + m;
        if (node < N) {
            float* op = out + (size_t)node * FEAT + l15;
            op[0]  = c0[r];
            op[16] = c1[r];
            op[32] = c2[r];
            op[48] = c3[r];
        }
    }
}

extern "C" void kernel_launch(void* const* d_in, const int* in_sizes, int n_in,
                              void* d_out, int out_size, void* d_ws, size_t ws_size,
                              hipStream_t stream) {
    const float* x  = (const float*)d_in[0];
    const int*   ei = (const int*)d_in[1];     // [2, E] row-major: row0=src, row1=dst
    const float* W  = (const float*)d_in[2];
    const int N = in_sizes[0] / FEAT;
    const int E = in_sizes[1] / 2;
    float* out = (float*)d_out;

    // Workspace layout: agg [N*64 f32] | cnt [N u32] | dis [N f32]
    float*    agg = (float*)d_ws;
    unsigned* cnt = (unsigned*)((char*)d_ws + (size_t)N * FEAT * sizeof(float));
    float*    dis = (float*)((char*)cnt + (size_t)N * sizeof(unsigned));

    // Zero agg + cnt (dis fully overwritten). Memset node is graph-capture safe.
    hipMemsetAsync(d_ws, 0, (size_t)N * (FEAT + 1) * sizeof(float), stream);

    gcn_deg_kernel<<<(E + 255) / 256, 256, 0, stream>>>(ei, cnt, E);
    gcn_dis_kernel<<<(N + 255) / 256, 256, 0, stream>>>(cnt, dis, N);

    const long long scat_threads = (long long)E * 16;
    gcn_scatter_kernel<<<(int)((scat_threads + 255) / 256), 256, 0, stream>>>(x, ei, dis, agg, E);

    gcn_gemm_kernel<<<(N + 31) / 32, 64, 0, stream>>>(agg, x, dis, W, out, N);
}